// SAGEConvMultiEdgeset_13589276524782
// MI455X (gfx1250) — compile-verified
//
#include <hip/hip_runtime.h>
#include <hip/hip_bf16.h>
#include <math.h>

// Problem constants from the reference
#define RC     4        // R*CH
#define NN     20000    // nodes
#define EE     100000   // edges
#define CC     256      // channels
#define MM     80000    // R*CH*N rows
#define EDIM   16       // edge_attr dim
#define LPITCH 260      // LDS pitch (pad 256 -> 260: bank = 4*row + k, conflict-free)

typedef float v2f __attribute__((ext_vector_type(2)));
typedef float v8f __attribute__((ext_vector_type(8)));
typedef int   b128_t __attribute__((vector_size(16)));   // 128-bit async payload

#if defined(__has_builtin)
#if __has_builtin(__builtin_amdgcn_global_load_async_to_lds_b128) && \
    __has_builtin(__builtin_amdgcn_s_wait_asynccnt)
#define HAVE_ASYNC_LDS 1
#endif
#endif

#define AS_GLOBAL __attribute__((address_space(1)))
#define AS_LDS    __attribute__((address_space(3)))

// ---------------------------------------------------------------------------
// Zero workspace (sums + counts are contiguous)
// ---------------------------------------------------------------------------
__global__ void zero_kernel(float* __restrict__ p, long n) {
    long i = (long)blockIdx.x * blockDim.x + threadIdx.x;
    if (i < n) p[i] = 0.0f;
}

// ---------------------------------------------------------------------------
// WMMA GEMM: C[M,256] = A0@B0 (+ A1@B1 if DUAL) + bias
// Block: 256 threads = 8 waves. Each wave owns a 32x16 tile (two 16x16 WMMA
// tiles sharing one B fragment -> half the global B loads per output).
// Block tile: 32 rows x 128 cols. grid = (M/32, 256/128).
// A tile (32x256 f32) staged in LDS via async-to-LDS loads (ASYNCcnt path);
// B streams from L2 (weights are 256KB, cache-resident) with a prefetch hint.
// ---------------------------------------------------------------------------
template <int DUAL>
__global__ void __launch_bounds__(256)
gemm_wmma_kernel(const float* __restrict__ A0,
                 const float* __restrict__ A1,
                 const float* __restrict__ B0,
                 const float* __restrict__ B1,
                 const float* __restrict__ bias,
                 float* __restrict__ Cout) {
    __shared__ float As[(DUAL ? 2 : 1) * 32 * LPITCH];

    const int row0 = blockIdx.x * 32;
    const int wave = threadIdx.x >> 5;   // 0..7 (wave32)
    const int lane = threadIdx.x & 31;
    const int col0 = blockIdx.y * 128 + wave * 16;
    const int t    = threadIdx.x;

    // Prefetch this block's B column strip (one line per K-row per thread).
    __builtin_prefetch(&B0[(size_t)t * CC + blockIdx.y * 128], 0, 1);
    if (DUAL) __builtin_prefetch(&B1[(size_t)t * CC + blockIdx.y * 128], 0, 1);

    // Stage A tile(s): 32x256 floats per source = 2048 float4, 8 per thread.
    for (int src = 0; src < (DUAL ? 2 : 1); ++src) {
        const float* A = src ? A1 : A0;
        float* dst = As + src * 32 * LPITCH;
#pragma unroll
        for (int i = 0; i < 8; ++i) {
            int idx = t + i * 256;          // 0..2047
            int r   = idx >> 6;             // 64 float4 per row
            int c4  = idx & 63;
#if defined(HAVE_ASYNC_LDS)
            __builtin_amdgcn_global_load_async_to_lds_b128(
                (AS_GLOBAL b128_t*)(A + (size_t)(row0 + r) * CC + c4 * 4),
                (AS_LDS    b128_t*)(dst + r * LPITCH + c4 * 4),
                0, 0);
#else
            float4 v = ((const float4*)(A + (size_t)(row0 + r) * CC))[c4];
            dst[r * LPITCH + c4 * 4 + 0] = v.x;
            dst[r * LPITCH + c4 * 4 + 1] = v.y;
            dst[r * LPITCH + c4 * 4 + 2] = v.z;
            dst[r * LPITCH + c4 * 4 + 3] = v.w;
#endif
        }
    }
#if defined(HAVE_ASYNC_LDS)
    __builtin_amdgcn_s_wait_asynccnt(0);
#endif
    __syncthreads();

    const int arow = lane & 15;     // M within 16-tile
    const int half = lane >> 4;     // selects K pair (A/B) / M+8 (C/D)
    const int bcol = col0 + arow;

    v8f acc0 = {};
    v8f acc1 = {};

    for (int src = 0; src < (DUAL ? 2 : 1); ++src) {
        const float* B    = src ? B1 : B0;
        const float* lds0 = As + src * 32 * LPITCH + arow * LPITCH;
        const float* lds1 = lds0 + 16 * LPITCH;

#pragma unroll 8
        for (int k0 = 0; k0 < CC; k0 += 4) {
            const int ka = k0 + 2 * half;
            v2f a0; a0.x = lds0[ka]; a0.y = lds0[ka + 1];
            v2f a1; a1.x = lds1[ka]; a1.y = lds1[ka + 1];
            v2f b;  b.x  = B[(size_t)ka * CC + bcol];
                    b.y  = B[(size_t)(ka + 1) * CC + bcol];
            acc0 = __builtin_amdgcn_wmma_f32_16x16x4_f32(
                false, a0, false, b, (short)0, acc0, false, false);
            acc1 = __builtin_amdgcn_wmma_f32_16x16x4_f32(
                false, a1, false, b, (short)0, acc1, false, false);
        }
    }

    const float bv = bias[bcol];
#pragma unroll
    for (int j = 0; j < 8; ++j) {
        // C/D layout: VGPR j -> M = j (lanes 0-15) / j+8 (lanes 16-31)
        Cout[(size_t)(row0 + j + 8 * half) * CC + bcol]      = acc0[j] + bv;
        Cout[(size_t)(row0 + 16 + j + 8 * half) * CC + bcol] = acc1[j] + bv;
    }
}

// ---------------------------------------------------------------------------
// Fused per-edge kernel: bond encoder + gather + exact GELU + weight +
// atomic scatter-add into sums (L2-resident), one count atomic per edge.
// One block (256 threads == 256 channels) per edge.
// ---------------------------------------------------------------------------
__global__ void __launch_bounds__(256)
edge_scatter_kernel(const float* __restrict__ x_l,        // [RC*N, C]
                    const float* __restrict__ edge_attr,  // [E, 16]
                    const float* __restrict__ edge_weight,// [RC, E]
                    const float* __restrict__ bond_W,     // [16, C]
                    const float* __restrict__ bond_b,     // [C]
                    const long long* __restrict__ edge_index, // [2, E]
                    float* __restrict__ sums,             // [RC*N, C]
                    float* __restrict__ cnt) {            // [N]
    const int e = blockIdx.x;
    const int c = threadIdx.x;

    __shared__ float attr[EDIM];
    __shared__ float wrc[RC];
    if (c < EDIM) attr[c] = edge_attr[(size_t)e * EDIM + c];
    if (c < RC)   wrc[c]  = edge_weight[(size_t)c * EE + e];

    const long long s = edge_index[e];
    const long long d = edge_index[EE + e];
    __syncthreads();

    // bond encoder for this edge/channel
    float emb = bond_b[c];
#pragma unroll
    for (int k = 0; k < EDIM; ++k)
        emb = fmaf(attr[k], bond_W[k * CC + c], emb);

#pragma unroll
    for (int rc = 0; rc < RC; ++rc) {
        float v = x_l[((size_t)rc * NN + (size_t)s) * CC + c] + emb;
        // exact GELU (erf form, matches torch F.gelu approximate='none')
        float g = 0.5f * v * (1.0f + erff(v * 0.70710678118654752f));
        atomicAdd(&sums[((size_t)rc * NN + (size_t)d) * CC + c], g * wrc[rc]);
    }
    if (c == 0) atomicAdd(&cnt[d], 1.0f);
}

// ---------------------------------------------------------------------------
// sums /= max(cnt, 1)  (in place -> becomes pooled features)
// ---------------------------------------------------------------------------
__global__ void __launch_bounds__(256)
div_kernel(float* __restrict__ sums, const float* __restrict__ cnt) {
    size_t i = (size_t)blockIdx.x * 256 + threadIdx.x;  // RC*N*C elements
    int n = (int)((i >> 8) % NN);
    sums[i] = sums[i] / fmaxf(cnt[n], 1.0f);
}

// ---------------------------------------------------------------------------
extern "C" void kernel_launch(void* const* d_in, const int* in_sizes, int n_in,
                              void* d_out, int out_size, void* d_ws, size_t ws_size,
                              hipStream_t stream) {
    (void)in_sizes; (void)n_in; (void)out_size; (void)ws_size;

    const float*     x           = (const float*)d_in[0];
    const float*     edge_attr   = (const float*)d_in[1];
    const float*     edge_weight = (const float*)d_in[2];
    const float*     lin_W       = (const float*)d_in[3];
    const float*     lin_b       = (const float*)d_in[4];
    const float*     linl_W      = (const float*)d_in[5];
    const float*     linl_b      = (const float*)d_in[6];
    const float*     linr_W      = (const float*)d_in[7];
    const float*     bond_W      = (const float*)d_in[8];
    const float*     bond_b      = (const float*)d_in[9];
    const long long* edge_index  = (const long long*)d_in[10]; // int64 per reference

    float* out = (float*)d_out;

    // Workspace layout (fp32): x_l [M*C] | sums [M*C] | cnt [N]  (~164 MB)
    float* x_l  = (float*)d_ws;
    float* sums = x_l + (size_t)MM * CC;
    float* cnt  = sums + (size_t)MM * CC;

    // 1) zero sums + cnt (contiguous)
    {
        long nz = (long)MM * CC + NN;
        zero_kernel<<<(unsigned)((nz + 255) / 256), 256, 0, stream>>>(sums, nz);
    }

    // 2) x_l = x @ lin_W + lin_b   (WMMA)
    gemm_wmma_kernel<0><<<dim3(MM / 32, CC / 128), 256, 0, stream>>>(
        x, nullptr, lin_W, nullptr, lin_b, x_l);

    // 3) fused bond-encoder + gather + GELU + weighted scatter-add
    edge_scatter_kernel<<<EE, 256, 0, stream>>>(
        x_l, edge_attr, edge_weight, bond_W, bond_b, edge_index, sums, cnt);

    // 4) mean
    div_kernel<<<RC * NN, 256, 0, stream>>>(sums, cnt);

    // 5) out = pooled @ linl_W + linl_b + x @ linr_W   (dual-A WMMA)
    gemm_wmma_kernel<1><<<dim3(MM / 32, CC / 128), 256, 0, stream>>>(
        sums, x, linl_W, linr_W, linl_b, out);
}